// GNNEncoder_73469710566065
// MI455X (gfx1250) — compile-verified
//
#include <hip/hip_runtime.h>

typedef __attribute__((ext_vector_type(2))) float v2f;
typedef __attribute__((ext_vector_type(8))) float v8f;

#define KDIM 128          // N_FEAT == N_HIDDEN == 128
#define ROWP 132          // LDS row pitch (floats): stride-4 banks -> conflict free, 8B aligned

// ---------------------------------------------------------------------------
// zero workspace (float4 stores; n must be a multiple of 4)
// ---------------------------------------------------------------------------
__global__ void zero_f32(float* __restrict__ p, long long n4) {
    long long i = (long long)blockIdx.x * blockDim.x + threadIdx.x;
    long long stride = (long long)gridDim.x * blockDim.x;
    float4* p4 = (float4*)p;
    float4 z = make_float4(0.f, 0.f, 0.f, 0.f);
    for (; i < n4; i += stride) p4[i] = z;
}

// ---------------------------------------------------------------------------
// One wave per edge: gather feat[src] row (float4/lane, coalesced 512B),
// scatter-add into sums[dst] with global_atomic_add_f32. Lane 0 optionally
// accumulates the in-degree count.
// ---------------------------------------------------------------------------
__global__ void scatter_mean_accum(const float4* __restrict__ feat,   // [N, 32] float4
                                   const int*    __restrict__ ei,     // [2, E]
                                   float*        __restrict__ sums,   // [N, 128]
                                   float*        __restrict__ cnt,    // [N] or null
                                   int nEdges) {
    int gid  = blockIdx.x * blockDim.x + threadIdx.x;
    int e    = gid >> 5;
    int lane = gid & 31;
    if (e >= nEdges) return;                 // wave-uniform
    int src = ei[e];
    int dst = ei[nEdges + e];
    float4 v = feat[(long long)src * 32 + lane];
    float* s = sums + (long long)dst * KDIM + lane * 4;
    atomicAdd(s + 0, v.x);
    atomicAdd(s + 1, v.y);
    atomicAdd(s + 2, v.z);
    atomicAdd(s + 3, v.w);
    if (cnt != nullptr && lane == 0) atomicAdd(&cnt[dst], 1.0f);
}

// ---------------------------------------------------------------------------
// Fused SAGEConv GEMM: out = act( (sums/max(cnt,1)) @ Wl + bias + xin @ Wr )
// K = 128 fixed, NOUT/RELU are compile-time so all B loads, LDS A loads and
// output stores use immediate offsets inside the fully unrolled K loop.
// Block = 256 threads = 8 waves; each block stages (8/tilesN)*16 A-rows of
// both input matrices into LDS (pre-scaled by 1/max(cnt,1)); each wave owns
// one 16x16 output tile: 32 K-steps x 2 chained V_WMMA_F32_16X16X4_F32.
// ---------------------------------------------------------------------------
template <int NOUT, bool RELU>
__global__ __launch_bounds__(256)
void sage_gemm_wmma(const float* __restrict__ sums,
                    const float* __restrict__ cnt,
                    const float* __restrict__ xin,
                    const float* __restrict__ Wl,    // [128, NOUT] row-major
                    const float* __restrict__ bias,  // [NOUT]
                    const float* __restrict__ Wr,    // [128, NOUT] row-major
                    float*       __restrict__ out) { // [N, NOUT]
    constexpr int TILESN = NOUT / 16;        // 8 (layer1) or 4 (layer2)
    constexpr int MPB    = 8 / TILESN;       // 1 or 2 M-tiles per block
    constexpr int ROWS   = MPB * 16;         // 16 or 32 staged rows

    __shared__ float sAgg[ROWS * ROWP];
    __shared__ float sX[ROWS * ROWP];
    __shared__ float sInv[ROWS];

    const int row0 = blockIdx.x * ROWS;
    const int tid  = threadIdx.x;

    if (tid < ROWS) {
        float c = cnt[row0 + tid];
        sInv[tid] = 1.0f / fmaxf(c, 1.0f);
    }
    __syncthreads();

    // Cooperative stage of A-rows (ROWS*32 float4 elements; uniform trip count)
#pragma unroll
    for (int i = tid; i < ROWS * (KDIM / 4); i += 256) {
        int r  = i >> 5;                    // staged row
        int k4 = (i & 31) * 4;              // k offset
        const float4 sv = *(const float4*)(sums + (long long)(row0 + r) * KDIM + k4);
        const float4 xv = *(const float4*)(xin  + (long long)(row0 + r) * KDIM + k4);
        float inv = sInv[r];
        float* da = sAgg + r * ROWP + k4;
        da[0] = sv.x * inv; da[1] = sv.y * inv; da[2] = sv.z * inv; da[3] = sv.w * inv;
        float* dx = sX + r * ROWP + k4;
        dx[0] = xv.x; dx[1] = xv.y; dx[2] = xv.z; dx[3] = xv.w;
    }
    __syncthreads();

    const int wave  = tid >> 5;
    const int lane  = tid & 31;
    const int tn    = wave % TILESN;        // N-tile owned by this wave
    const int msub  = wave / TILESN;        // M-sub-tile within block
    const int mrow  = lane & 15;            // M row (A) / N col (B,C,D)
    const int khalf = lane >> 4;            // K-half select per ISA layout
    const int colB  = tn * 16 + mrow;

    // Hoisted per-lane bases: all loop offsets below are compile-time imms.
    const float* aAggRow = sAgg + (msub * 16 + mrow) * ROWP + 2 * khalf;
    const float* aXRow   = sX   + (msub * 16 + mrow) * ROWP + 2 * khalf;
    const float* WlBase  = Wl + colB + 2 * khalf * NOUT;
    const float* WrBase  = Wr + colB + 2 * khalf * NOUT;

    v8f acc = {};
#pragma unroll
    for (int k0 = 0; k0 < KDIM; k0 += 4) {
        // A tiles: lane l<16 -> {K=k0,k0+1}, lane l>=16 -> {K=k0+2,k0+3}
        v2f aA = *(const v2f*)(aAggRow + k0);
        v2f aX = *(const v2f*)(aXRow + k0);
        // B tiles: same K striping, N = lane&15 across lanes (coalesced)
        v2f bL, bR;
        bL.x = WlBase[k0 * NOUT];
        bL.y = WlBase[k0 * NOUT + NOUT];
        bR.x = WrBase[k0 * NOUT];
        bR.y = WrBase[k0 * NOUT + NOUT];
        acc = __builtin_amdgcn_wmma_f32_16x16x4_f32(false, aA, false, bL,
                                                    (short)0, acc, false, false);
        acc = __builtin_amdgcn_wmma_f32_16x16x4_f32(false, aX, false, bR,
                                                    (short)0, acc, false, false);
    }

    const float b = bias[colB];
    const int rbase = row0 + msub * 16 + 8 * khalf;   // C/D: VGPR j -> M=j(+8)
    float* outBase = out + (long long)rbase * NOUT + colB;
#pragma unroll
    for (int j = 0; j < 8; ++j) {
        float v = acc[j] + b;
        if (RELU) v = fmaxf(v, 0.0f);
        outBase[j * NOUT] = v;               // immediate offsets, clause-able
    }
}

// ---------------------------------------------------------------------------
extern "C" void kernel_launch(void* const* d_in, const int* in_sizes, int n_in,
                              void* d_out, int out_size, void* d_ws, size_t ws_size,
                              hipStream_t stream) {
    const float* x   = (const float*)d_in[0];
    const float* Wl1 = (const float*)d_in[1];
    const float* bl1 = (const float*)d_in[2];
    const float* Wr1 = (const float*)d_in[3];
    const float* Wl2 = (const float*)d_in[4];
    const float* bl2 = (const float*)d_in[5];
    const float* Wr2 = (const float*)d_in[6];
    const int*   ei  = (const int*)d_in[7];

    const int N = in_sizes[0] / KDIM;       // 100000
    const int E = in_sizes[7] / 2;          // 1600000

    // Workspace layout (floats): cnt[N] (padded) | sums[N*128] | h[N*128]
    float* ws      = (float*)d_ws;
    const long long cntPad = 102400;        // 256B-aligned region for cnt
    float* cnt  = ws;
    float* sums = ws + cntPad;
    float* h    = sums + (long long)N * KDIM;
    float* out  = (float*)d_out;

    const long long zn1 = cntPad + (long long)N * KDIM;   // cnt + sums
    zero_f32<<<2048, 256, 0, stream>>>(ws, zn1 / 4);

    const int scatterBlocks = (E * 32 + 255) / 256;
    // Layer 1 aggregation (+ in-degree)
    scatter_mean_accum<<<scatterBlocks, 256, 0, stream>>>((const float4*)x, ei,
                                                          sums, cnt, E);
    // h = relu(agg1 @ W_l1 + b_l1 + x @ W_r1) ; NOUT=128 -> 1 M-tile/block
    sage_gemm_wmma<128, true><<<N / 16, 256, 0, stream>>>(sums, cnt, x, Wl1,
                                                          bl1, Wr1, h);
    // Re-zero sums for layer 2
    zero_f32<<<2048, 256, 0, stream>>>(sums, ((long long)N * KDIM) / 4);
    // Layer 2 aggregation over h
    scatter_mean_accum<<<scatterBlocks, 256, 0, stream>>>((const float4*)h, ei,
                                                          sums, nullptr, E);
    // out = agg2 @ W_l2 + b_l2 + h @ W_r2 ; NOUT=64 -> 2 M-tiles/block
    sage_gemm_wmma<64, false><<<N / 32, 256, 0, stream>>>(sums, cnt, h, Wl2,
                                                          bl2, Wr2, out);
}